// LinearMimo_40742059769933
// MI455X (gfx1250) — compile-verified
//
#include <hip/hip_runtime.h>

// MIMO IIR filter bank: y[b,t,o] = sum_i IIR_{o,i}(u[b,:,i])[t]
// B=16, T=4096, I=32, O=32, NB=3, NA=2.
//
// Chunked-scan formulation (L=16): within a chunk the response is a 16x16
// lower-triangular Toeplitz matmul (v_wmma_f32_16x16x4_f32 x4) over raw u,
// plus rank-1 boundary terms from the carry state (y[t0-1], y[t0-2]) and the
// FIR tail (u[t0-1], u[t0-2]). One wave per (o,i) pair; N dimension = batch.
// Reduction over i is amortized: each wave produces GROUP=4 chunk tiles
// between workgroup barriers (carry propagates wave-locally via ds_bpermute),
// then all 1024 threads reduce 4 tiles' worth of elements at once.

typedef __attribute__((ext_vector_type(2))) float v2f;
typedef __attribute__((ext_vector_type(8))) float v8f;

#define Tn 4096
#define Bn 16
#define In 32
#define On 32
#define Ln 16
#define NC (Tn / Ln)     // 256 chunks
#define GROUP 4          // chunks per barrier/reduction round
#define NG (NC / GROUP)  // 64 groups

// ---------------------------------------------------------------------------
// Pass 1: transpose u[b][t][i] -> ut[i][t][b] so each chunk's B-tile
// (16 timesteps x 16 batches for one channel i) is 1KB contiguous.
// ---------------------------------------------------------------------------
__global__ __launch_bounds__(512) void u_transpose_kernel(
    const float* __restrict__ u, float* __restrict__ ut) {
  __shared__ float tile[512];
  const int t = blockIdx.x;
  const int tid = threadIdx.x;
  {
    const int b = tid >> 5, i = tid & 31;          // coalesced read over i
    tile[i * 16 + b] = u[(b * Tn + t) * In + i];
  }
  __syncthreads();
  {
    const int i = tid >> 4, b = tid & 15;          // coalesced write over b
    ut[(i * Tn + t) * Bn + b] = tile[i * 16 + b];
  }
}

// ---------------------------------------------------------------------------
// Pass 2: chunked IIR scan. grid = O (32 WGs), block = 1024 (32 waves),
// wave w handles pair (o = blockIdx.x, i = w). Deterministic LDS reduction
// over i, direct stores to d_out (no atomics anywhere).
// ---------------------------------------------------------------------------
__global__ __launch_bounds__(1024) void mimo_iir_wmma_kernel(
    const float* __restrict__ ut,
    const float* __restrict__ bc,   // (O, I, 3)
    const float* __restrict__ ac,   // (O, I, 2)
    float* __restrict__ out) {      // (B, T, O)
  __shared__ float hsh[In][20];                // per-wave impulse response h[0..17]
  __shared__ float wsh[In][16];                // per-wave combined kernel w[0..15]
  __shared__ float wtile[In * GROUP * 256];    // 128 KB: [i][cc][n*16+m]

  const int o    = blockIdx.x;
  const int wave = threadIdx.x >> 5;     // = input channel i
  const int lane = threadIdx.x & 31;
  const int hi   = lane >> 4;            // half-wave select
  const int n    = lane & 15;            // N index (batch) / A-row M

  // Per-pair coefficients (uniform within the wave).
  const float b0 = bc[(o * In + wave) * 3 + 0];
  const float b1 = bc[(o * In + wave) * 3 + 1];
  const float b2 = bc[(o * In + wave) * 3 + 2];
  const float a1 = ac[(o * In + wave) * 2 + 0];
  const float a2 = ac[(o * In + wave) * 2 + 1];

  // Impulse response h and combined FIR*IIR kernel w, staged in LDS so every
  // lane can gather the entries it needs (lane-dependent indices).
  if (lane == 0) {
    float hm1 = 0.0f, hm2 = 0.0f;  // h[m-1], h[m-2]
    for (int m = 0; m < 18; ++m) {
      const float hn = (m == 0) ? 1.0f : (-a1 * hm1 - a2 * hm2);
      hsh[wave][m] = hn;
      if (m < 16) wsh[wave][m] = b0 * hn + b1 * hm1 + b2 * hm2;
      hm2 = hm1;
      hm1 = hn;
    }
  }
  __syncthreads();

  // A fragments: A[M][sigma] = w[M - sigma] (lower-triangular Toeplitz).
  // Assumed 16x16x4 f32 layout: lane L -> M = L%16, K = 2*(L/16) + v.
  v2f afrag[4];
#pragma unroll
  for (int k = 0; k < 4; ++k) {
#pragma unroll
    for (int v = 0; v < 2; ++v) {
      const int sigma = 4 * k + 2 * hi + v;
      const int m = n - sigma;
      afrag[k][v] = (m >= 0) ? wsh[wave][m] : 0.0f;
    }
  }

  // Boundary-term coefficients, loop-invariant. D-tile rows for this lane are
  // M = r + 8*hi; we need h[M-1], h[M], h[M+1]  ->  harr[j] = h[8*hi - 1 + j].
  float harr[10];
#pragma unroll
  for (int j = 0; j < 10; ++j) {
    const int idx = 8 * hi - 1 + j;
    harr[j] = (idx >= 0) ? hsh[wave][idx] : 0.0f;
  }
  __syncthreads();

  const float* gbase = ut + wave * (Tn * Bn);  // this wave's channel slice
  float yp1 = 0.0f, yp2 = 0.0f;                // y[t0-1], y[t0-2] for batch n

  for (int g = 0; g < NG; ++g) {
    // -- GROUP chunks back-to-back, no barriers: loads of chunk cc+1 pipeline
    //    under the WMMA chain of chunk cc.
#pragma unroll
    for (int cc = 0; cc < GROUP; ++cc) {
      const int c = g * GROUP + cc;
      const float* p = gbase + c * (Ln * Bn) + n;

      // B fragments: raw u, K = local timestep sigma, N = batch.
      v2f bfrag[4];
#pragma unroll
      for (int k = 0; k < 4; ++k) {
#pragma unroll
        for (int v = 0; v < 2; ++v)
          bfrag[k][v] = p[(4 * k + 2 * hi + v) * Bn];
      }

      // Previous-chunk u samples for the FIR tail (zero before t=0).
      float um1 = 0.0f, um2 = 0.0f;
      if (c > 0) {
        um1 = p[-Bn];
        um2 = p[-2 * Bn];
      }

      // Seed C with the rank-1 boundary terms:
      //   y[M] += h[M+1]*yp1 + h[M]*(-a2*yp2 + b1*um1 + b2*um2) + h[M-1]*(b2*um1)
      const float s0 = b1 * um1 + b2 * um2 - a2 * yp2;
      const float s1 = b2 * um1;
      v8f acc;
#pragma unroll
      for (int r = 0; r < 8; ++r)
        acc[r] = harr[r + 2] * yp1 + harr[r + 1] * s0 + harr[r] * s1;

      // 16x16x16 fp32 matmul = 4 chained K=4 WMMAs.
      acc = __builtin_amdgcn_wmma_f32_16x16x4_f32(false, afrag[0], false, bfrag[0],
                                                  (short)0, acc, false, false);
      acc = __builtin_amdgcn_wmma_f32_16x16x4_f32(false, afrag[1], false, bfrag[1],
                                                  (short)0, acc, false, false);
      acc = __builtin_amdgcn_wmma_f32_16x16x4_f32(false, afrag[2], false, bfrag[2],
                                                  (short)0, acc, false, false);
      acc = __builtin_amdgcn_wmma_f32_16x16x4_f32(false, afrag[3], false, bfrag[3],
                                                  (short)0, acc, false, false);

      // Carry update: rows M=15,14 live in VGPR 7,6 of the hi=1 half
      // (16x16 f32 C/D layout). Broadcast to all lanes with the same batch n.
      yp1 = __shfl(acc[7], 16 + n, 32);
      yp2 = __shfl(acc[6], 16 + n, 32);

      // Stash this chunk's tile in its LDS slot (two b128 stores per lane).
      float4* dst = (float4*)&wtile[(wave * GROUP + cc) * 256 + n * 16 + 8 * hi];
      dst[0] = make_float4(acc[0], acc[1], acc[2], acc[3]);
      dst[1] = make_float4(acc[4], acc[5], acc[6], acc[7]);
    }

    // Prefetch next group's 4KB slice of this channel into the cache so the
    // post-barrier loads don't stall the carry chain.
    if (g + 1 < NG)
      __builtin_prefetch(gbase + (g + 1) * (GROUP * Ln * Bn) + lane * 32, 0, 3);

    __syncthreads();

    // -- Deterministic reduction over i: all 1024 threads, fixed order.
    {
      const int e = threadIdx.x;           // (cc, elem) = (e>>8, e&255)
      float s = 0.0f;
#pragma unroll
      for (int w = 0; w < In; ++w) s += wtile[w * (GROUP * 256) + e];
      const int cc   = e >> 8;
      const int elem = e & 255;
      const int bb   = elem >> 4;          // batch
      const int m    = elem & 15;          // local timestep
      const int t    = g * (GROUP * Ln) + cc * Ln + m;
      out[(bb * Tn + t) * On + o] = s;
    }
    __syncthreads();  // protect wtile before next group's stores
  }
}

// ---------------------------------------------------------------------------
extern "C" void kernel_launch(void* const* d_in, const int* in_sizes, int n_in,
                              void* d_out, int out_size, void* d_ws, size_t ws_size,
                              hipStream_t stream) {
  (void)in_sizes; (void)n_in; (void)out_size; (void)ws_size;
  const float* u  = (const float*)d_in[0];  // (B, T, I) f32
  const float* bc = (const float*)d_in[1];  // (O, I, 3) f32
  const float* ac = (const float*)d_in[2];  // (O, I, 2) f32
  float* out = (float*)d_out;               // (B, T, O) f32
  float* ut  = (float*)d_ws;                // (I, T, B) f32 = 8 MB scratch

  u_transpose_kernel<<<Tn, 512, 0, stream>>>(u, ut);
  mimo_iir_wmma_kernel<<<On, 1024, 0, stream>>>(ut, bc, ac, out);
}